// HardNegativeTripletLoss_87316685128547
// MI455X (gfx1250) — compile-verified
//
#include <hip/hip_runtime.h>
#include <hip/hip_bf16.h>
#include <math.h>

typedef __attribute__((ext_vector_type(2))) float v2f;
typedef __attribute__((ext_vector_type(8))) float v8f;

#define NB      8192
#define ND      512
#define NTILES  (NB / 16)      // 512 column tiles of 16
#define NCHUNK  16             // grid.y: column-chunk count
#define JT_PER_CHUNK (NTILES / NCHUNK)   // 32 tiles per chunk
#define LDS_PAD 516            // 512 + 4 floats: conflict-free ds_load_b64

// ---------------------------------------------------------------------------
// Kernel 1: row squared norms + sentinel init for the atomic max/min arrays.
// ---------------------------------------------------------------------------
__global__ __launch_bounds__(256) void norms_init_kernel(
    const float* __restrict__ X, float* __restrict__ sq,
    unsigned* __restrict__ hposBits, unsigned* __restrict__ hnegBits)
{
    const int lane = threadIdx.x & 31;
    const int wave = threadIdx.x >> 5;
    const int row  = blockIdx.x * 8 + wave;            // grid = 1024 blocks
    const float* xr = X + (size_t)row * ND;

    float s = 0.0f;
    #pragma unroll
    for (int c = lane * 4; c < ND; c += 32 * 4) {
        const float4 v = *(const float4*)(xr + c);
        s = fmaf(v.x, v.x, s); s = fmaf(v.y, v.y, s);
        s = fmaf(v.z, v.z, s); s = fmaf(v.w, v.w, s);
    }
    #pragma unroll
    for (int off = 16; off; off >>= 1) s += __shfl_xor(s, off, 32);

    if (lane == 0) {
        sq[row]       = s;
        hposBits[row] = 0xFF800000u;   // -inf : max-of-d2 sentinel
        hnegBits[row] = 0x7F800000u;   // +inf : min-of-d2 sentinel
    }
}

// ---------------------------------------------------------------------------
// Kernel 2: fused tiled distance + hard mining on SQUARED distances.
// sqrt is monotonic, so argmax/argmin over d == over d2; sqrt moves to the
// finalize kernel. Epilogue is branchless selects + fmax/fmin only.
// ---------------------------------------------------------------------------
__global__ __launch_bounds__(32) void triplet_tile_kernel(
    const float* __restrict__ X, const int* __restrict__ labels,
    const float* __restrict__ sq,
    unsigned* __restrict__ hposBits, unsigned* __restrict__ hnegBits)
{
    __shared__ float As[16][LDS_PAD];

    const int lane = threadIdx.x;          // 0..31
    const int I0   = blockIdx.x * 16;      // row strip base
    const int jt0  = blockIdx.y * JT_PER_CHUNK;

    // --- stage the 16x512 A strip into LDS (float4 per lane per iter) ---
    for (int t = lane; t < 16 * (ND / 4); t += 32) {
        const int r = t >> 7;              // / (ND/4)
        const int c = (t & 127) << 2;
        const float4 v = *(const float4*)(X + (size_t)(I0 + r) * ND + c);
        *(float4*)&As[r][c] = v;
    }
    __syncthreads();

    const int half = lane >> 4;            // 0 or 1
    const int nloc = lane & 15;            // N within tile / M for A-frag

    // rows covered by this lane's accumulator slots: m = half*8 + r
    float sqi[8]; int labi[8];
    #pragma unroll
    for (int r = 0; r < 8; ++r) {
        const int gi = I0 + half * 8 + r;
        sqi[r]  = sq[gi];
        labi[r] = labels[gi];
    }

    float hp2[8], hn2[8];                  // running max/min of squared dist
    #pragma unroll
    for (int r = 0; r < 8; ++r) { hp2[r] = -INFINITY; hn2[r] = INFINITY; }

    // A-fragment base: lane holds A[m=nloc][k0 + half*2 .. +1]  (16x4 f32 layout)
    const float* aPtr = &As[nloc][half * 2];

    for (int jt = 0; jt < JT_PER_CHUNK; ++jt) {
        const int j0 = (jt0 + jt) * 16;
        const int gj = j0 + nloc;
        // B-fragment: lane holds B[k0 + half*2 .. +1][n=nloc] = X[gj][k...]
        const float* bPtr = X + (size_t)gj * ND + half * 2;

        v8f acc = {};
        #pragma unroll 8
        for (int k0 = 0; k0 < ND; k0 += 4) {
            const v2f a = *(const v2f*)(aPtr + k0);   // ds_load_b64, conflict-free
            const v2f b = *(const v2f*)(bPtr + k0);   // global_load_b64, L2 hit
            acc = __builtin_amdgcn_wmma_f32_16x16x4_f32(
                false, a, false, b, (short)0, acc, false, false);
        }

        const float sqj  = sq[gj];
        const int   labj = labels[gj];
        #pragma unroll
        for (int r = 0; r < 8; ++r) {
            const int gi = I0 + half * 8 + r;
            float d2 = fmaf(-2.0f, acc[r], sqi[r] + sqj);
            d2 = fmaxf(d2, 0.0f);
            const bool same = (labi[r] == labj);
            const float candP = (same && (gi != gj)) ? d2 : -INFINITY;
            const float candN = same ? INFINITY : d2;
            hp2[r] = fmaxf(hp2[r], candP);
            hn2[r] = fminf(hn2[r], candN);
        }
    }

    // reduce over the 16 lanes of each half (they hold different n, same m set)
    #pragma unroll
    for (int r = 0; r < 8; ++r) {
        #pragma unroll
        for (int off = 8; off; off >>= 1) {
            hp2[r] = fmaxf(hp2[r], __shfl_xor(hp2[r], off, 32));
            hn2[r] = fminf(hn2[r], __shfl_xor(hn2[r], off, 32));
        }
    }

    // merge across column chunks. Clamped d2 >= 0 (plus +/-inf sentinels),
    // so IEEE bit patterns order identically to signed ints.
    if (nloc == 0) {
        #pragma unroll
        for (int r = 0; r < 8; ++r) {
            const int gi = I0 + half * 8 + r;
            atomicMax((int*)&hposBits[gi], __float_as_int(hp2[r]));
            atomicMin((int*)&hnegBits[gi], __float_as_int(hn2[r]));
        }
    }
}

// ---------------------------------------------------------------------------
// Kernel 3: final loss reduction over the 8192 rows (sqrt applied here).
// ---------------------------------------------------------------------------
__global__ __launch_bounds__(256) void finalize_kernel(
    const unsigned* __restrict__ hposBits, const unsigned* __restrict__ hnegBits,
    float* __restrict__ out)
{
    __shared__ float sLoss[256];
    __shared__ float sCnt[256];

    float loss = 0.0f, cnt = 0.0f;
    for (int i = threadIdx.x; i < NB; i += 256) {
        const float p2 = __int_as_float((int)hposBits[i]);
        const float n2 = __int_as_float((int)hnegBits[i]);
        const bool valid = (p2 > -INFINITY) && (n2 < INFINITY);
        if (valid) {
            const float p = sqrtf(p2 + 1e-16f);
            const float n = sqrtf(n2 + 1e-16f);
            loss += fmaxf(p - n + 0.3f, 0.0f);
            cnt  += 1.0f;
        }
    }
    sLoss[threadIdx.x] = loss;
    sCnt[threadIdx.x]  = cnt;
    __syncthreads();
    for (int s = 128; s; s >>= 1) {
        if (threadIdx.x < s) {
            sLoss[threadIdx.x] += sLoss[threadIdx.x + s];
            sCnt[threadIdx.x]  += sCnt[threadIdx.x + s];
        }
        __syncthreads();
    }
    if (threadIdx.x == 0) out[0] = sLoss[0] / fmaxf(sCnt[0], 1.0f);
}

// ---------------------------------------------------------------------------
extern "C" void kernel_launch(void* const* d_in, const int* in_sizes, int n_in,
                              void* d_out, int out_size, void* d_ws, size_t ws_size,
                              hipStream_t stream) {
    const float* X      = (const float*)d_in[0];   // [8192,512] fp32
    const int*   labels = (const int*)d_in[1];     // [8192]

    float*    sq  = (float*)d_ws;                              // 32 KB
    unsigned* hpb = (unsigned*)((char*)d_ws + (size_t)NB * 4); // 32 KB
    unsigned* hnb = (unsigned*)((char*)d_ws + (size_t)NB * 8); // 32 KB

    norms_init_kernel<<<NB / 8, 256, 0, stream>>>(X, sq, hpb, hnb);

    dim3 grid(NTILES, NCHUNK);   // 512 row strips x 16 column chunks
    triplet_tile_kernel<<<grid, 32, 0, stream>>>(X, labels, sq, hpb, hnb);

    finalize_kernel<<<1, 256, 0, stream>>>(hpb, hnb, (float*)d_out);
}